// ScaledDotProductAttention_22256520528798
// MI455X (gfx1250) — compile-verified
//
#include <hip/hip_runtime.h>

typedef __attribute__((ext_vector_type(2))) float v2f;
typedef __attribute__((ext_vector_type(4))) float f4;
typedef __attribute__((ext_vector_type(8))) float v8f;

#define B_ 8
#define N_ 6
#define D_ 256
#define CHW 524288            // 512*1024
#define OUT_MAIN (8*6*524288) // main output elements; attn^T appended after
#define INV_T 0.0625f         // 1/TEMPERATURE

// ---------------------------------------------------------------------------
// Kernel 1: scores via V_WMMA_F32_16X16X4_F32 + lane-local sparsemax.
// One wave per batch. A = k (M=key rows), B = q^T (N=query cols), K=256 in
// 64 steps of 4. Padding lanes (mn >= 6) load clamped row-0 data
// UNCONDITIONALLY: their garbage only lands in D rows/cols 6..15, which are
// never read — so no EXEC predication, no zero-fill, tight load+load+wmma
// inner loop with EXEC all-ones throughout (as WMMA requires).
// D layout: lane n (<16) holds D[m][n] for m=0..7 in c[0..7]; with only 6
// keys, lane q (<6) owns the entire score column for query q -> sparsemax is
// lane-local (no shuffles, no LDS).
// ---------------------------------------------------------------------------
__global__ __launch_bounds__(32) void attn_sparsemax_kernel(
    const float* __restrict__ q, const float* __restrict__ k,
    float* __restrict__ attnT)
{
  const int b    = blockIdx.x;
  const int lane = threadIdx.x;   // 0..31
  const int mn   = lane & 15;     // M for A-frag, N for B-frag
  const int h    = lane >> 4;     // K-half selector (K = 2h, 2h+1)
  const int row  = (mn < N_) ? mn : 0;   // clamp padding lanes to row 0
  const float* krow = k + (size_t)(b * N_ + row) * D_ + 2 * h;
  const float* qrow = q + (size_t)(b * N_ + row) * D_ + 2 * h;

  v8f c = {};
  #pragma unroll 8
  for (int d0 = 0; d0 < D_; d0 += 4) {
    v2f a  = *(const v2f*)(krow + d0);
    v2f bb = *(const v2f*)(qrow + d0);
    // D = A(16x4 f32) * B(4x16 f32) + C
    c = __builtin_amdgcn_wmma_f32_16x16x4_f32(
        /*neg_a=*/false, a, /*neg_b=*/false, bb,
        /*c_mod=*/(short)0, c, /*reuse_a=*/false, /*reuse_b=*/false);
  }

  if (lane < N_) {   // lane == query index; c[0..5] = raw scores over 6 keys
    float z[N_], zs[N_];
    #pragma unroll
    for (int j = 0; j < N_; ++j) { z[j] = c[j] * INV_T; zs[j] = z[j]; }
    // insertion sort, descending (6 elements)
    #pragma unroll
    for (int i = 1; i < N_; ++i) {
      float key = zs[i]; int j = i - 1;
      while (j >= 0 && zs[j] < key) { zs[j + 1] = zs[j]; --j; }
      zs[j + 1] = key;
    }
    // support size + tau (Martins & Astudillo 2016)
    float cum = 0.0f, zc_k = 0.0f; int ksup = 1;
    #pragma unroll
    for (int i = 0; i < N_; ++i) {
      cum += zs[i];
      if (1.0f + (float)(i + 1) * zs[i] > cum) { ksup = i + 1; zc_k = cum; }
    }
    const float tau = (zc_k - 1.0f) / (float)ksup;
    float* wout = attnT + (size_t)(b * N_ + lane) * N_;  // attn^T[b][q][k]
    #pragma unroll
    for (int j = 0; j < N_; ++j) wout[j] = fmaxf(z[j] - tau, 0.0f);
  }
}

// ---------------------------------------------------------------------------
// Kernel 2: out[b][q][i] = sum_k w[b][q][k] * v[b][k][i], i in [0, 524288).
// HBM-bound streaming (~201 MB total traffic -> ~8.7 us floor at 23.3 TB/s):
// each v element read exactly once (reused across all 6 queries), float4
// (B128) non-temporal loads/stores. 512 blocks per batch -> b uniform per
// block, so the 36 weights become scalar loads.
// ---------------------------------------------------------------------------
__global__ __launch_bounds__(256) void weighted_v_kernel(
    const float* __restrict__ v, const float* __restrict__ w,
    float* __restrict__ out)
{
  const int tile = blockIdx.x;          // 4096 tiles total, 512 per batch
  const int b    = tile >> 9;           // uniform within block
  const size_t i = ((size_t)(tile & 511) * 256 + threadIdx.x) * 4;

  const float* wb = w + b * 36;         // attn^T[b]: [q][k]
  float wl[36];
  #pragma unroll
  for (int t = 0; t < 36; ++t) wl[t] = wb[t];

  const size_t base = (size_t)b * N_ * CHW + i;
  f4 vv[N_];
  #pragma unroll
  for (int kk = 0; kk < N_; ++kk)
    vv[kk] = __builtin_nontemporal_load((const f4*)(v + base + (size_t)kk * CHW));

  #pragma unroll
  for (int qq = 0; qq < N_; ++qq) {
    f4 acc = vv[0] * wl[qq * 6 + 0];
    #pragma unroll
    for (int kk = 1; kk < N_; ++kk) acc += vv[kk] * wl[qq * 6 + kk];
    __builtin_nontemporal_store(acc, (f4*)(out + base + (size_t)qq * CHW));
  }
}

extern "C" void kernel_launch(void* const* d_in, const int* in_sizes, int n_in,
                              void* d_out, int out_size, void* d_ws, size_t ws_size,
                              hipStream_t stream) {
  (void)in_sizes; (void)n_in; (void)out_size; (void)d_ws; (void)ws_size;
  const float* q = (const float*)d_in[0];
  const float* k = (const float*)d_in[1];
  const float* v = (const float*)d_in[2];
  float* out   = (float*)d_out;
  float* attnT = out + OUT_MAIN;        // second output, also weights for K2

  attn_sparsemax_kernel<<<B_, 32, 0, stream>>>(q, k, attnT);
  weighted_v_kernel<<<B_ * 512, 256, 0, stream>>>(v, attnT, out);
}